// NTXentLoss_4045859193248
// MI455X (gfx1250) — compile-verified
//
#include <hip/hip_runtime.h>
#include <hip/hip_bf16.h>
#include <stdint.h>

// ---------------------------------------------------------------------------
// NT-Xent loss, fused: normalize -> (zn znT)/T tiles via V_WMMA_F32_16X16X4_F32
// -> streaming softmax denominator (fixed max shift 2.0) -> mean NLL.
// B tiles are DMA'd into LDS with the CDNA5 Tensor Data Mover (double-buffered,
// pad_enable gives a 258-float LDS row stride => conflict-free ds_load_b64).
// ---------------------------------------------------------------------------

#define NB    4096      // batch
#define NROW  8192      // 2B rows
#define DD    256       // feature dim
#define NT    512       // 16-wide column tiles over 8192 columns
#define NSLICE 4        // column slices (grid.y)
#define TILES_PER_SLICE (NT / NSLICE)   // 128

typedef float v2f __attribute__((ext_vector_type(2)));
typedef float v8f __attribute__((ext_vector_type(8)));
typedef unsigned int u32x4 __attribute__((ext_vector_type(4)));
typedef int i32x4 __attribute__((ext_vector_type(4)));
typedef int i32x8 __attribute__((ext_vector_type(8)));

// ---------------------------------------------------------------------------
// Kernel 1: row-normalize z = [z_i; z_j], fold sqrt(1/T)=sqrt(2) into the data
// so that (s*zn)(s*zn)^T = logits directly.  One wave per row.
// ---------------------------------------------------------------------------
__global__ void __launch_bounds__(256)
ntxent_normalize(const float* __restrict__ zi, const float* __restrict__ zj,
                 float* __restrict__ zn) {
    const int wave = threadIdx.x >> 5;
    const int lane = threadIdx.x & 31;
    const int row  = blockIdx.x * 8 + wave;
    const float* src = (row < NB) ? (zi + (size_t)row * DD)
                                  : (zj + (size_t)(row - NB) * DD);
    float v[8];
    float ss = 0.0f;
#pragma unroll
    for (int i = 0; i < 8; ++i) {
        v[i] = src[lane + 32 * i];
        ss += v[i] * v[i];
    }
#pragma unroll
    for (int m = 1; m < 32; m <<= 1)
        ss += __shfl_xor(ss, m, 32);
    const float norm  = sqrtf(ss);
    const float scale = 1.41421356237309515f / fmaxf(norm, 1e-8f);  // sqrt(2)/max(||z||,eps)
    float* dst = zn + (size_t)row * DD;
#pragma unroll
    for (int i = 0; i < 8; ++i)
        dst[lane + 32 * i] = v[i] * scale;
}

// ---------------------------------------------------------------------------
// TDM: load one contiguous 16x256-f32 tile (16 KB) of zn into LDS with
// 2-dword padding after every 256 dwords (row stride 258 floats).
// Descriptor per CDNA5 ISA ch.8 (D# group0/group1); groups 2/3 unused (zero).
// 6-arg builtin form: (g0 u32x4, g1 i32x8, g2 i32x4, g3 i32x4, i32x8, cpol).
// ---------------------------------------------------------------------------
__device__ __forceinline__ void tdm_load_tile(const float* gsrc, unsigned ldsByteOff) {
    const unsigned long long ga = (unsigned long long)(uintptr_t)gsrc;
    u32x4 g0;
    g0[0] = 1u;                                        // count=1 (valid), is_restore=0
    g0[1] = ldsByteOff;                                // lds_addr [63:32]
    g0[2] = (unsigned)(ga & 0xFFFFFFFFu);              // global_addr [95:64]
    g0[3] = (unsigned)((ga >> 32) & 0x01FFFFFFu)       // global_addr [120:96]
          | 0x80000000u;                               // type=2 ("image") [127:126]
    i32x8 g1;
    g1[0] = 0x03D20000;        // data_size=4B(2)<<16 | pad_enable | pad_interval=7(256dw)
                               // | pad_amount=1(2dw); workgroup_mask=0
    g1[1] = (int)(4096u << 16);// tensor_dim0 = 4096 (low 16 bits at [63:48])
    g1[2] = (int)(512u << 16); // tensor_dim0 hi=0 | tensor_dim1 = 512
    g1[3] = (int)(4096u << 16);// tensor_dim1 hi=0 | tile_dim0 = 4096 elements
    g1[4] = 1;                 // tile_dim1 = 1, tile_dim2 = 0
    g1[5] = 4096;              // tensor_dim0_stride (low 32)
    g1[6] = 0;
    g1[7] = 0;
    const i32x4 gz4 = {0, 0, 0, 0};
    const i32x8 gz8 = {0, 0, 0, 0, 0, 0, 0, 0};
    __builtin_amdgcn_tensor_load_to_lds(g0, g1, gz4, gz4, gz8, 0);
}

// ---------------------------------------------------------------------------
// Kernel 2: fused GEMM + streaming exp-sum + target-logit extraction.
// grid = (64 row-blocks, 4 column-slices), block = 256 (8 waves).
// Wave w owns rows [bx*128 + w*16, +16); its A strip (16x256) sits in 128 VGPRs.
// All 8 waves share double-buffered TDM-loaded B tiles in LDS.
// ---------------------------------------------------------------------------
#define LDS_ROW 258
#define LDS_TILE (16 * LDS_ROW)   // 4128 floats per buffer

__global__ void __launch_bounds__(256)
ntxent_main(const float* __restrict__ zn, float* __restrict__ s_part,
            float* __restrict__ tgt) {
    __shared__ __align__(16) float ldsB[2][LDS_TILE];

    const int wave   = threadIdx.x >> 5;
    const int lane   = threadIdx.x & 31;
    const int lane15 = lane & 15;
    const int h      = lane >> 4;           // K-pair half select
    const int slice  = blockIdx.y;
    const int r0     = blockIdx.x * 128 + wave * 16;

    // ---- A strip: 16 rows x 256 K in registers (float2 x 64 = 128 VGPRs) ----
    const float* arow = zn + (size_t)(r0 + lane15) * DD + 2 * h;
    v2f a[64];
#pragma unroll
    for (int kk = 0; kk < 64; ++kk)
        a[kk] = *(const v2f*)(arow + 4 * kk);

    // Column tile holding this wave's target (diagonal) elements.
    const int ct_target = (r0 & (NB - 1)) >> 4;

    float s[8];
#pragma unroll
    for (int k = 0; k < 8; ++k) s[k] = 0.0f;

    const int ct0 = slice * TILES_PER_SLICE;

    // Prologue: DMA first B tile of this slice.
    if (wave == 0)
        tdm_load_tile(zn + (size_t)ct0 * (16 * DD),
                      (unsigned)(uintptr_t)&ldsB[0][0]);

    for (int t = 0; t < TILES_PER_SLICE; ++t) {
        const int ct  = ct0 + t;
        const int buf = t & 1;

        if (wave == 0) {
            if (t + 1 < TILES_PER_SLICE) {
                tdm_load_tile(zn + (size_t)(ct + 1) * (16 * DD),
                              (unsigned)(uintptr_t)&ldsB[buf ^ 1][0]);
                __builtin_amdgcn_s_wait_tensorcnt(1);  // tile t landed
            } else {
                __builtin_amdgcn_s_wait_tensorcnt(0);
            }
        }
        __syncthreads();   // B tile t visible to all waves

        // ---- 16x16 logits tile via 64 chained V_WMMA_F32_16X16X4_F32 ----
        v8f c = {0.f, 0.f, 0.f, 0.f, 0.f, 0.f, 0.f, 0.f};
        const float* brow = &ldsB[buf][lane15 * LDS_ROW + 2 * h];
#pragma unroll
        for (int kk = 0; kk < 64; ++kk) {
            v2f b = *(const v2f*)(brow + 4 * kk);
            c = __builtin_amdgcn_wmma_f32_16x16x4_f32(
                    false, a[kk], false, b, (short)0, c, false, false);
        }

        // ---- target (label) logit: diagonal element (M == N) of owning tile ----
        if (ct == ct_target) {
#pragma unroll
            for (int k = 0; k < 8; ++k) {
                if (h == 0 && lane15 == k)     tgt[r0 + k]     = c[k];
                if (h == 1 && lane15 == k + 8) tgt[r0 + 8 + k] = c[k];
            }
        }

        // ---- streaming softmax denominator; logits <= 2.0 => fixed shift ----
#pragma unroll
        for (int k = 0; k < 8; ++k)
            s[k] += __expf(c[k] - 2.0f);

        __syncthreads();   // everyone done reading buf before TDM reuses it
    }

    // ---- combine the 16 column-lanes of each half-wave (rows fixed/lane) ----
#pragma unroll
    for (int m = 1; m < 16; m <<= 1) {
#pragma unroll
        for (int k = 0; k < 8; ++k)
            s[k] += __shfl_xor(s[k], m, 32);
    }
    // lane 0 -> rows r0..r0+7, lane 16 -> rows r0+8..r0+15
    if (lane15 == 0) {
        float* dst = s_part + (size_t)slice * NROW + r0 + 8 * h;
#pragma unroll
        for (int k = 0; k < 8; ++k)
            dst[k] = s[k];
    }
}

// ---------------------------------------------------------------------------
// Kernel 3: nll_r = 2 + log(sum_slices s) - tgt_r ; out = mean(nll).
// Single block, deterministic tree reduction (no float atomics).
// ---------------------------------------------------------------------------
__global__ void __launch_bounds__(256)
ntxent_finalize(const float* __restrict__ s_part, const float* __restrict__ tgt,
                float* __restrict__ out) {
    __shared__ float red[256];
    const int tid = threadIdx.x;
    float acc = 0.0f;
    for (int r = tid; r < NROW; r += 256) {
        float s = s_part[r] + s_part[NROW + r] + s_part[2 * NROW + r] + s_part[3 * NROW + r];
        acc += 2.0f + logf(s) - tgt[r];
    }
    red[tid] = acc;
    __syncthreads();
    for (int off = 128; off > 0; off >>= 1) {
        if (tid < off) red[tid] += red[tid + off];
        __syncthreads();
    }
    if (tid == 0) out[0] = red[0] * (1.0f / (float)NROW);
}

// ---------------------------------------------------------------------------
// Host-side launcher.
// Workspace layout: zn [8192*256 f32] | s_part [4*8192 f32] | tgt [8192 f32]
// = 8.16 MB total.
// ---------------------------------------------------------------------------
extern "C" void kernel_launch(void* const* d_in, const int* in_sizes, int n_in,
                              void* d_out, int out_size, void* d_ws, size_t ws_size,
                              hipStream_t stream) {
    const float* zi = (const float*)d_in[0];
    const float* zj = (const float*)d_in[1];
    float* out = (float*)d_out;

    float* zn     = (float*)d_ws;                       // 8192*256
    float* s_part = zn + (size_t)NROW * DD;             // 4*8192
    float* tgt    = s_part + (size_t)NSLICE * NROW;     // 8192

    ntxent_normalize<<<NROW / 8, 256, 0, stream>>>(zi, zj, zn);
    ntxent_main<<<dim3(NROW / 128, NSLICE), 256, 0, stream>>>(zn, s_part, tgt);
    ntxent_finalize<<<1, 256, 0, stream>>>(s_part, tgt, out);
}